// HGT_44856638439786
// MI455X (gfx1250) — compile-verified
//
#include <hip/hip_runtime.h>
#include <math.h>

// ---------------- problem constants ----------------
#define NN      10000
#define NE      320000
#define FDIM    128
#define HEADS   8
#define DHEAD   16
#define NVIEWS  8
#define NLAYERS 2
#define MTILES  625          // NN / 16

typedef float v2f __attribute__((ext_vector_type(2)));
typedef float v8f __attribute__((ext_vector_type(8)));

// ---------------- small device helpers ----------------
__device__ __forceinline__ float geluf(float x) {
    // jax.nn.gelu default (approximate=True, tanh form)
    const float c = 0.7978845608028654f;
    return 0.5f * x * (1.0f + tanhf(c * (x + 0.044715f * x * x * x)));
}
__device__ __forceinline__ float sigmoidf(float x) { return 1.0f / (1.0f + expf(-x)); }

// monotonic float <-> uint mapping for atomicMax-based segment max
__device__ __forceinline__ unsigned enc_f32(float f) {
    unsigned u = __float_as_uint(f);
    return (u & 0x80000000u) ? ~u : (u | 0x80000000u);
}
__device__ __forceinline__ float dec_f32(unsigned k) {
    return (k & 0x80000000u) ? __uint_as_float(k & 0x7FFFFFFFu) : __uint_as_float(~k);
}
#define ENC_NEG_INF 0x007FFFFFu   // enc(-inf)

// ---------------- fill kernels ----------------
__global__ void fill_f32_kernel(float* p, float v, int n) {
    int i = blockIdx.x * blockDim.x + threadIdx.x;
    if (i < n) p[i] = v;
}
__global__ void fill_u32_kernel(unsigned* p, unsigned v, int n) {
    int i = blockIdx.x * blockDim.x + threadIdx.x;
    if (i < n) p[i] = v;
}

// ---------------- tiny softmax over a short vector (wn / wt) ----------------
__global__ void softmax_vec_kernel(const float* __restrict__ x, float* y, int n) {
    if (threadIdx.x == 0 && blockIdx.x == 0) {
        float m = -3.402823466e38f;
        for (int i = 0; i < n; ++i) m = fmaxf(m, x[i]);
        float s = 0.f;
        for (int i = 0; i < n; ++i) s += expf(x[i] - m);
        for (int i = 0; i < n; ++i) y[i] = expf(x[i] - m) / s;
    }
}

// ---------------- WMMA f32 GEMM: C[M,128] = epi(A[M,128] @ W[128,128] + b) ----------------
// One wave computes a full 16x128 row strip: 8 accumulator fragments, A fragment
// loaded once per K-step and reused across all 8 column tiles (8 WMMAs per step).
// EPI: 0 plain, 1 relu, 2 skip-mix (s*acc + (1-s)*mixPrev, s=sigmoid(*scal)), 3 C += (*scal)*acc
// GELU_A: apply gelu to A elements on load.
template <int EPI, bool GELU_A>
__global__ __launch_bounds__(128) void gemm128_kernel(
    const float* __restrict__ A, const float* __restrict__ W,
    const float* __restrict__ bias, float* C,
    const float* mixPrev, const float* __restrict__ scal, int M)
{
    const int wave = threadIdx.x >> 5;
    const int lane = threadIdx.x & 31;
    const int r  = lane & 15;
    const int hh = lane >> 4;          // which half of the wave
    const int mt = blockIdx.x * 4 + wave;
    if (mt * 16 >= M) return;          // wave-uniform
    const int m0 = mt * 16;

    const float* Arow = A + (size_t)(m0 + r) * FDIM;
    v8f c[8];
    #pragma unroll
    for (int nt = 0; nt < 8; ++nt) c[nt] = (v8f){};

    for (int kk = 0; kk < FDIM; kk += 4) {
        const int ka = kk + 2 * hh;
        float ax = Arow[ka], ay = Arow[ka + 1];
        if (GELU_A) { ax = geluf(ax); ay = geluf(ay); }
        v2f a; a.x = ax; a.y = ay;
        const float* W0 = W + (size_t)ka * FDIM + r;        // row ka
        const float* W1 = W0 + FDIM;                        // row ka+1
        #pragma unroll
        for (int nt = 0; nt < 8; ++nt) {
            v2f b; b.x = W0[nt * 16]; b.y = W1[nt * 16];
            c[nt] = __builtin_amdgcn_wmma_f32_16x16x4_f32(false, a, false, b, (short)0, c[nt], false, false);
        }
    }

    float s = 0.f;
    if (EPI == 2) s = sigmoidf(scal[0]);
    if (EPI == 3) s = scal[0];
    #pragma unroll
    for (int nt = 0; nt < 8; ++nt) {
        #pragma unroll
        for (int v = 0; v < 8; ++v) {
            const int m = m0 + v + 8 * hh;
            const int n = nt * 16 + r;
            const size_t off = (size_t)m * FDIM + n;
            float acc = c[nt][v] + bias[n];
            if (EPI == 0)      C[off] = acc;
            else if (EPI == 1) C[off] = acc > 0.f ? acc : 0.f;
            else if (EPI == 2) C[off] = s * acc + (1.f - s) * mixPrev[off];
            else               C[off] += s * acc;
        }
    }
}

// ---------------- A = sigmoid(Z @ Z^T), [10000 x 10000] ----------------
// One wave computes a 16x64 tile (4 column tiles): A fragment reused 4x per K-step.
#define ZGROUPS 157   // ceil(625 / 4) column-tile groups of 4
__global__ __launch_bounds__(256) void zzt_sigmoid_kernel(
    const float* __restrict__ Z, float* __restrict__ Aout)
{
    const int wave = threadIdx.x >> 5;
    const int lane = threadIdx.x & 31;
    const int r  = lane & 15;
    const int hh = lane >> 4;
    const int mt = blockIdx.x;                 // 0..624
    const int g  = blockIdx.y * 8 + wave;      // column-tile group
    if (g >= ZGROUPS) return;                  // wave-uniform
    const int m0 = mt * 16;
    const int nt0 = g * 4;

    const float* Arow = Z + (size_t)(m0 + r) * FDIM;
    const float* Brow[4];
    v8f c[4];
    #pragma unroll
    for (int s = 0; s < 4; ++s) {
        c[s] = (v8f){};
        const int nt = (nt0 + s < MTILES) ? (nt0 + s) : (MTILES - 1);  // clamp (extra work, safe)
        Brow[s] = Z + (size_t)(nt * 16 + r) * FDIM;                    // B[k][n] = Z[n][k]
    }

    for (int kk = 0; kk < FDIM; kk += 4) {
        const int ka = kk + 2 * hh;
        v2f a; a.x = Arow[ka]; a.y = Arow[ka + 1];
        #pragma unroll
        for (int s = 0; s < 4; ++s) {
            v2f b; b.x = Brow[s][ka]; b.y = Brow[s][ka + 1];
            c[s] = __builtin_amdgcn_wmma_f32_16x16x4_f32(false, a, false, b, (short)0, c[s], false, false);
        }
    }
    #pragma unroll
    for (int s = 0; s < 4; ++s) {
        if (nt0 + s >= MTILES) break;          // wave-uniform
        #pragma unroll
        for (int v = 0; v < 8; ++v) {
            const int m = m0 + v + 8 * hh;
            const int n = (nt0 + s) * 16 + r;
            Aout[(size_t)m * NN + n] = sigmoidf(c[s][v]);
        }
    }
}

// ---------------- per-head relation transform: Y[n,h,:] = X[n,h,:] @ Wrel[h] ----------------
__global__ void rel_transform_kernel(const float* __restrict__ X,
                                     const float* __restrict__ Wrel,
                                     float* __restrict__ Y)
{
    int idx = blockIdx.x * blockDim.x + threadIdx.x;
    if (idx >= NN * FDIM) return;
    const int e = idx & 15;
    const int h = (idx >> 4) & 7;
    const int n = idx >> 7;
    const float* xr = X + (size_t)n * FDIM + h * DHEAD;
    const float* w  = Wrel + h * (DHEAD * DHEAD) + e;   // w[d*16 + e]
    float s = 0.f;
    #pragma unroll
    for (int d = 0; d < DHEAD; ++d) s += xr[d] * w[d * DHEAD];
    Y[idx] = s;
}

// ---------------- attention score + segment max ----------------
__global__ void alpha_score_kernel(const float* __restrict__ q,
                                   const float* __restrict__ kr,
                                   const int* __restrict__ edges,
                                   const float* __restrict__ prel,
                                   float* __restrict__ alphaE,
                                   unsigned* __restrict__ mkey)
{
    int idx = blockIdx.x * blockDim.x + threadIdx.x;
    if (idx >= NE * HEADS) return;
    const int h = idx & 7;
    const int e = idx >> 3;
    const int src = edges[e];
    const int dst = edges[NE + e];
    const float* qr = q  + (size_t)dst * FDIM + h * DHEAD;
    const float* kp = kr + (size_t)src * FDIM + h * DHEAD;
    float s = 0.f;
    #pragma unroll
    for (int d = 0; d < DHEAD; ++d) s += qr[d] * kp[d];
    s *= prel[h] * 0.25f;                 // scale = 1/sqrt(16)
    alphaE[idx] = s;
    atomicMax(&mkey[dst * HEADS + h], enc_f32(s));
}

// ---------------- exp(score - max) + segment sum ----------------
__global__ void alpha_expsum_kernel(float* __restrict__ alphaE,
                                    const int* __restrict__ edges,
                                    const unsigned* __restrict__ mkey,
                                    float* __restrict__ ssum)
{
    int idx = blockIdx.x * blockDim.x + threadIdx.x;
    if (idx >= NE * HEADS) return;
    const int h = idx & 7;
    const int e = idx >> 3;
    const int dst = edges[NE + e];
    const float m = dec_f32(mkey[dst * HEADS + h]);
    const float v = expf(alphaE[idx] - m);
    alphaE[idx] = v;
    atomicAdd(&ssum[dst * HEADS + h], v);
}

// ---------------- normalized message scatter: out[dst] += vr[src] * alpha ----------------
__global__ void scatter_msg_kernel(const float* __restrict__ alphaE,
                                   const float* __restrict__ ssum,
                                   const float* __restrict__ vr,
                                   const int* __restrict__ edges,
                                   float* __restrict__ outacc)
{
    int idx = blockIdx.x * blockDim.x + threadIdx.x;
    if (idx >= NE * HEADS) return;
    const int h = idx & 7;
    const int e = idx >> 3;
    const int src = edges[e];
    const int dst = edges[NE + e];
    const float al = alphaE[idx] / (ssum[dst * HEADS + h] + 1e-16f);
    const float* vp = vr + (size_t)src * FDIM + h * DHEAD;
    float* op = outacc + (size_t)dst * FDIM + h * DHEAD;
    #pragma unroll
    for (int d = 0; d < DHEAD; ++d) atomicAdd(&op[d], vp[d] * al);
}

// ---------------- t = (X @ t_w + t_b) * wt[c]  -> traw [NN,2] ----------------
__global__ void trow_kernel(const float* __restrict__ X, const float* __restrict__ tw,
                            const float* __restrict__ tb, const float* __restrict__ wt,
                            float* __restrict__ traw)
{
    int idx = blockIdx.x * blockDim.x + threadIdx.x;
    if (idx >= NN * 2) return;
    const int c = idx & 1;
    const int r = idx >> 1;
    const float* xr = X + (size_t)r * FDIM;
    float s = tb[c];
    #pragma unroll 8
    for (int k = 0; k < FDIM; ++k) s += xr[k] * tw[k * 2 + c];
    traw[idx] = s * wt[c];
}

// ---------------- column-wise softmax over 10000 rows (2 columns) ----------------
__global__ __launch_bounds__(256) void col_softmax_kernel(const float* __restrict__ traw,
                                                          float* __restrict__ T)
{
    __shared__ float red[256];
    const int tid = threadIdx.x;
    for (int c = 0; c < 2; ++c) {
        float m = -3.402823466e38f;
        for (int r = tid; r < NN; r += 256) m = fmaxf(m, traw[r * 2 + c]);
        red[tid] = m; __syncthreads();
        for (int s = 128; s > 0; s >>= 1) { if (tid < s) red[tid] = fmaxf(red[tid], red[tid + s]); __syncthreads(); }
        m = red[0]; __syncthreads();
        float sum = 0.f;
        for (int r = tid; r < NN; r += 256) sum += expf(traw[r * 2 + c] - m);
        red[tid] = sum; __syncthreads();
        for (int s = 128; s > 0; s >>= 1) { if (tid < s) red[tid] += red[tid + s]; __syncthreads(); }
        sum = red[0]; __syncthreads();
        for (int r = tid; r < NN; r += 256) T[r * 2 + c] = expf(traw[r * 2 + c] - m) / sum;
        __syncthreads();
    }
}

// ================= host-side orchestration =================
extern "C" void kernel_launch(void* const* d_in, const int* in_sizes, int n_in,
                              void* d_out, int out_size, void* d_ws, size_t ws_size,
                              hipStream_t stream)
{
    (void)in_sizes; (void)n_in; (void)out_size; (void)ws_size;
    // inputs (setup_inputs dict order)
    const float* x_views = (const float*)d_in[0];   // [8,NN,128]
    const int*   e_fol   = (const int*)  d_in[1];   // [2,NE]
    const int*   e_fri   = (const int*)  d_in[2];   // [2,NE]
    const float* lin_w   = (const float*)d_in[3];   // [8,128,128]
    const float* lin_b   = (const float*)d_in[4];   // [8,128]
    const float* k_w     = (const float*)d_in[5];   // [2,128,128]
    const float* k_b     = (const float*)d_in[6];
    const float* q_w     = (const float*)d_in[7];
    const float* q_b     = (const float*)d_in[8];
    const float* v_w     = (const float*)d_in[9];
    const float* v_b     = (const float*)d_in[10];
    const float* a_w     = (const float*)d_in[11];
    const float* a_b     = (const float*)d_in[12];
    const float* skip    = (const float*)d_in[13];  // [2]
    const float* a_rel   = (const float*)d_in[14];  // [2,2,8,16,16]
    const float* m_rel   = (const float*)d_in[15];
    const float* p_rel   = (const float*)d_in[16];  // [2,2,8]
    const float* out_w   = (const float*)d_in[17];
    const float* out_b   = (const float*)d_in[18];
    const float* recon_w = (const float*)d_in[19];
    const float* recon_b = (const float*)d_in[20];
    const float* t_w     = (const float*)d_in[21];  // [128,2]
    const float* t_b     = (const float*)d_in[22];  // [2]
    const float* weight  = (const float*)d_in[23];  // [8]
    const float* wnodety = (const float*)d_in[24];  // [2]

    // outputs (concatenated flat): A, X, T, wn, wt
    float* out       = (float*)d_out;
    float* A_out     = out;                                      // 1e8
    float* X_out     = out + (size_t)NN * NN;                    // 1.28e6
    float* T_out     = X_out + (size_t)NN * FDIM;                // 2e4
    float* wn_out    = T_out + (size_t)NN * 2;                   // 8
    float* wt_out    = wn_out + NVIEWS;                          // 2

    // workspace layout (floats)
    float* ws = (float*)d_ws;
    const size_t NF = (size_t)NN * FDIM;     // 1,280,000
    float*    Z      = ws;
    float*    h      = ws + 1 * NF;
    float*    kbuf   = ws + 2 * NF;
    float*    qbuf   = ws + 3 * NF;
    float*    vbuf   = ws + 4 * NF;
    float*    krbuf  = ws + 5 * NF;
    float*    vrbuf  = ws + 6 * NF;
    float*    outacc = ws + 7 * NF;
    float*    alphaE = ws + 8 * NF;                         // NE*8 = 2,560,000
    unsigned* mkey   = (unsigned*)(ws + 8 * NF + (size_t)NE * HEADS);
    float*    ssum   = (float*)(mkey + (size_t)NN * HEADS);
    float*    traw   = ssum + (size_t)NN * HEADS;           // NN*2

    const int EHB = (NE * HEADS + 255) / 256;   // 10000 blocks
    const int NFB = ((int)NF + 255) / 256;      // 5000 blocks
    const int NHB = (NN * HEADS + 255) / 256;   // 313 blocks
    const int GEMMB = (MTILES + 3) / 4;         // 157 blocks of 4 waves

    // fusion weights (needed by later GEMM epilogues)
    softmax_vec_kernel<<<1, 32, 0, stream>>>(weight,  wn_out, NVIEWS);
    softmax_vec_kernel<<<1, 32, 0, stream>>>(wnodety, wt_out, 2);

    // Z = 0
    fill_f32_kernel<<<NFB, 256, 0, stream>>>(Z, 0.f, (int)NF);

    for (int view = 0; view < NVIEWS; ++view) {
        // h = relu(x @ lin_w + lin_b)
        gemm128_kernel<1, false><<<GEMMB, 128, 0, stream>>>(
            x_views + (size_t)view * NF, lin_w + (size_t)view * FDIM * FDIM,
            lin_b + (size_t)view * FDIM, h, nullptr, nullptr, NN);

        for (int l = 0; l < NLAYERS; ++l) {
            const size_t wo = (size_t)l * FDIM * FDIM;
            const size_t bo = (size_t)l * FDIM;
            gemm128_kernel<0, false><<<GEMMB, 128, 0, stream>>>(h, k_w + wo, k_b + bo, kbuf, nullptr, nullptr, NN);
            gemm128_kernel<0, false><<<GEMMB, 128, 0, stream>>>(h, q_w + wo, q_b + bo, qbuf, nullptr, nullptr, NN);
            gemm128_kernel<0, false><<<GEMMB, 128, 0, stream>>>(h, v_w + wo, v_b + bo, vbuf, nullptr, nullptr, NN);

            fill_f32_kernel<<<NFB, 256, 0, stream>>>(outacc, 0.f, (int)NF);

            for (int et = 0; et < 2; ++et) {
                const int* edges = (et == 0) ? e_fol : e_fri;
                const size_t ro = (size_t)(l * 2 + et) * HEADS * DHEAD * DHEAD;
                const size_t po = (size_t)(l * 2 + et) * HEADS;

                rel_transform_kernel<<<NFB, 256, 0, stream>>>(kbuf, a_rel + ro, krbuf);
                rel_transform_kernel<<<NFB, 256, 0, stream>>>(vbuf, m_rel + ro, vrbuf);

                fill_u32_kernel<<<NHB, 256, 0, stream>>>(mkey, ENC_NEG_INF, NN * HEADS);
                fill_f32_kernel<<<NHB, 256, 0, stream>>>(ssum, 0.f, NN * HEADS);

                alpha_score_kernel <<<EHB, 256, 0, stream>>>(qbuf, krbuf, edges, p_rel + po, alphaE, mkey);
                alpha_expsum_kernel<<<EHB, 256, 0, stream>>>(alphaE, edges, mkey, ssum);
                scatter_msg_kernel <<<EHB, 256, 0, stream>>>(alphaE, ssum, vrbuf, edges, outacc);
            }

            // h = s*(gelu(outacc) @ a_w + a_b) + (1-s)*h,  s = sigmoid(skip[l])
            gemm128_kernel<2, true><<<GEMMB, 128, 0, stream>>>(
                outacc, a_w + wo, a_b + bo, h, h, skip + l, NN);
        }

        // Z += wn[view] * (h @ out_w + out_b)
        gemm128_kernel<3, false><<<GEMMB, 128, 0, stream>>>(
            h, out_w, out_b, Z, nullptr, wn_out + view, NN);
    }

    // X = Z @ recon_w + recon_b
    gemm128_kernel<0, false><<<GEMMB, 128, 0, stream>>>(Z, recon_w, recon_b, X_out, nullptr, nullptr, NN);

    // T = softmax_axis0((X @ t_w + t_b) * wt)
    trow_kernel<<<(NN * 2 + 255) / 256, 256, 0, stream>>>(X_out, t_w, t_b, wt_out, traw);
    col_softmax_kernel<<<1, 256, 0, stream>>>(traw, T_out);

    // A = sigmoid(Z @ Z^T)
    dim3 zgrid(MTILES, (ZGROUPS + 7) / 8);
    zzt_sigmoid_kernel<<<zgrid, 256, 0, stream>>>(Z, A_out);
}